// ALiBiAttention_12713103196714
// MI455X (gfx1250) — compile-verified
//
#include <hip/hip_runtime.h>

// ---------------------------------------------------------------------------
// ALiBi multi-head attention, fp32, CDNA5 (gfx1250).
// WMMA f32 16x16x4 matrix path + async global->LDS staging (ASYNCcnt),
// double-buffered LDS tiles. B=2, S=2048, D=512, H=8, Hd=64.
// ---------------------------------------------------------------------------

typedef float v2f __attribute__((ext_vector_type(2)));
typedef float v8f __attribute__((ext_vector_type(8)));

#define WMMA_F32(a, b, c) \
  __builtin_amdgcn_wmma_f32_16x16x4_f32(false, (a), false, (b), (short)0, (c), false, false)

#define TS 64   // GEMM tile (M, N, K block)
#define HD 64   // head dim
#define KB 64   // keys staged per LDS panel in attention

// --- CDNA5 async global->LDS copy helpers (ASYNCcnt-tracked) ---------------
__device__ __forceinline__ unsigned lds_off(const void* p) {
  // low 32 bits of a generic pointer to LDS == LDS byte address
  return (unsigned)(size_t)p;
}
__device__ __forceinline__ void async_b128(unsigned dst_lds, const float* src) {
  asm volatile("global_load_async_to_lds_b128 %0, %1, off"
               :: "v"(dst_lds), "v"(src) : "memory");
}
__device__ __forceinline__ void wait_async_16() {   // 16 ops still in flight OK
  asm volatile("s_wait_asynccnt 0x10" ::: "memory");
}
__device__ __forceinline__ void wait_async_0() {    // all async ops done
  asm volatile("s_wait_asynccnt 0x0" ::: "memory");
}

// ---------------------------------------------------------------------------
// Generic GEMM: out[m][j] = sum_k in[m][k] * W[j][k] + bias[j]
// inHeadMajor : in  is [B,H,S,Hd] with m=b*S+s, k=h*Hd+d
// outHeadMajor: out is [B,H,S,Hd] with m=b*S+s, j=h*Hd+d
// 128 threads = 4 waves; wave computes 16x64 slice; double-buffered async LDS.
// ---------------------------------------------------------------------------
__global__ __launch_bounds__(128) void gemm_wmma_f32(
    const float* __restrict__ in, const float* __restrict__ W,
    const float* __restrict__ bias, float* __restrict__ out,
    int M, int N, int K, int inHeadMajor, int outHeadMajor,
    int S, int Hd, int H)
{
  __shared__ __align__(16) float xs[2][TS][TS + 4];  // row stride 272B (16B mult)
  __shared__ __align__(16) float ws[2][TS][TS + 4];

  const int tid  = threadIdx.x;
  const int lane = tid & 31;
  const int wave = tid >> 5;
  const int mBlock = blockIdx.x * TS;
  const int nBlock = blockIdx.y * TS;
  const int nIdx = lane & 15;   // A-row / B-col / C-col index
  const int hi   = lane >> 4;   // upper/lower half of wave
  const int kb   = 2 * hi;      // k base within a 4-wide WMMA step

  // stage one 64x64 in-tile + 64x64 W-tile: 16 async B128 copies per thread
  auto stage = [&](int buf, int k0) {
    for (int i = tid; i < TS * (TS / 4); i += 128) {
      int r = i >> 4, c4 = i & 15;
      int m = mBlock + r;
      const float* src;
      if (inHeadMajor) {
        int b = m / S, s = m % S;
        int k = k0 + c4 * 4;
        int h = k / Hd, d = k % Hd;
        src = in + (((size_t)(b * H + h) * S + s) * Hd + d);
      } else {
        src = in + ((size_t)m * K + k0 + c4 * 4);
      }
      async_b128(lds_off(&xs[buf][r][c4 * 4]), src);
      async_b128(lds_off(&ws[buf][r][c4 * 4]),
                 W + (size_t)(nBlock + r) * K + k0 + c4 * 4);
    }
  };

  v8f acc[4] = {};

  stage(0, 0);
  const int nIter = K / TS;
  for (int it = 0; it < nIter; ++it) {
    const int buf = it & 1;
    if (it + 1 < nIter) { stage(buf ^ 1, (it + 1) * TS); wait_async_16(); }
    else                { wait_async_0(); }
    __syncthreads();

    const int mw = wave * 16;
    for (int kk = 0; kk < TS; kk += 4) {
      v2f a;
      a.x = xs[buf][mw + nIdx][kk + kb];
      a.y = xs[buf][mw + nIdx][kk + kb + 1];
#pragma unroll
      for (int nt = 0; nt < 4; nt++) {
        v2f b;
        b.x = ws[buf][nt * 16 + nIdx][kk + kb];
        b.y = ws[buf][nt * 16 + nIdx][kk + kb + 1];
        acc[nt] = WMMA_F32(a, b, acc[nt]);
      }
    }
    __syncthreads();   // everyone done reading buf before it is re-staged
  }

  // epilogue: add bias, scatter per layout.  C/D: elem v = C[v+8*hi][lane&15]
#pragma unroll
  for (int nt = 0; nt < 4; nt++) {
    int j = nBlock + nt * 16 + nIdx;
    float bv = bias[j];
#pragma unroll
    for (int v = 0; v < 8; v++) {
      int m = mBlock + wave * 16 + v + 8 * hi;
      float val = acc[nt][v] + bv;
      if (outHeadMajor) {
        int b = m / S, s = m % S;
        int h = j / Hd, d = j % Hd;
        out[((size_t)(b * H + h) * S + s) * Hd + d] = val;
      } else {
        out[(size_t)m * N + j] = val;
      }
    }
  }
}

// ---------------------------------------------------------------------------
// Flash attention with ALiBi + padding mask.
// Block = 128 threads = 4 waves, 64 queries of one (b,h).
// Q pre-scaled, in registers; K/V panels double-buffered via async copies;
// online softmax; P relaid C/D->A via per-wave LDS scratch (s_wait_dscnt).
// ---------------------------------------------------------------------------
__global__ __launch_bounds__(128) void attn_wmma_f32(
    const float* __restrict__ Q, const float* __restrict__ Kt,
    const float* __restrict__ V, const unsigned char* __restrict__ mask,
    float* __restrict__ O, int S, int H)
{
  __shared__ __align__(16) float Ks[2][KB][HD + 4];
  __shared__ __align__(16) float Vs[2][KB][HD + 4];
  __shared__ float Ps[4][16][17];

  const int tid  = threadIdx.x;
  const int lane = tid & 31;
  const int wave = tid >> 5;
  const int bh = blockIdx.y;               // b*H + h
  const int b  = bh / H;
  const int h  = bh % H;
  const int q0 = blockIdx.x * 64 + wave * 16;
  const int nIdx = lane & 15;
  const int hi   = lane >> 4;
  const int kb   = 2 * hi;

  const float* Qh = Q  + (size_t)bh * S * HD;
  const float* Kh = Kt + (size_t)bh * S * HD;
  const float* Vh = V  + (size_t)bh * S * HD;
  const float slope = exp2f(-(float)(h + 1));   // H=8 power-of-two slopes
  const float scale = 0.125f;                   // 1/sqrt(64)

  // stage one 64-key K panel + V panel: 16 async B128 copies per thread
  auto stageKV = [&](int buf, int kblk) {
    for (int i = tid; i < KB * (HD / 4); i += 128) {
      int r = i >> 4, c4 = i & 15;
      async_b128(lds_off(&Ks[buf][r][c4 * 4]),
                 Kh + (size_t)(kblk + r) * HD + c4 * 4);
      async_b128(lds_off(&Vs[buf][r][c4 * 4]),
                 Vh + (size_t)(kblk + r) * HD + c4 * 4);
    }
  };

  // preload Q A-fragments (16 k-steps over Hd=64), pre-scaled
  v2f aQ[16];
#pragma unroll
  for (int t = 0; t < 16; t++) {
    const float* qp = Qh + (size_t)(q0 + nIdx) * HD + 4 * t + kb;
    aQ[t].x = qp[0] * scale;
    aQ[t].y = qp[1] * scale;
  }

  v8f accO[4] = {};
  float mrow[8], lrow[8];
#pragma unroll
  for (int v = 0; v < 8; v++) { mrow[v] = -3.0e38f; lrow[v] = 0.0f; }

  stageKV(0, 0);
  const int nPanels = S / KB;
  for (int it = 0; it < nPanels; ++it) {
    const int buf  = it & 1;
    const int kblk = it * KB;
    if (it + 1 < nPanels) { stageKV(buf ^ 1, kblk + KB); wait_async_16(); }
    else                  { wait_async_0(); }
    __syncthreads();

    for (int sub = 0; sub < KB; sub += 16) {
      const int k0 = kblk + sub;

      // ---- scores: 16x16 = Q(16x64) . K^T, via 16 chained f32 WMMAs ----
      v8f accS = {};
#pragma unroll
      for (int t = 0; t < 16; t++) {
        v2f bK;
        bK.x = Ks[buf][sub + nIdx][4 * t + kb];
        bK.y = Ks[buf][sub + nIdx][4 * t + kb + 1];
        accS = WMMA_F32(aQ[t], bK, accS);
      }

      // ---- ALiBi bias + padding mask (C/D layout registers) ----
      const int kIdx = k0 + nIdx;
      const float colBias = mask[(size_t)b * S + kIdx] ? 0.0f : -3.0e38f;
      float sv[8], alpha[8], pv[8];
#pragma unroll
      for (int v = 0; v < 8; v++) {
        int qIdx = q0 + v + 8 * hi;
        sv[v] = accS[v] - slope * fabsf((float)(qIdx - kIdx)) + colBias;
      }

      // ---- online softmax: row reductions across 16 lanes of the half ----
#pragma unroll
      for (int v = 0; v < 8; v++) {
        float mx = sv[v];
        mx = fmaxf(mx, __shfl_xor(mx, 1, 32));
        mx = fmaxf(mx, __shfl_xor(mx, 2, 32));
        mx = fmaxf(mx, __shfl_xor(mx, 4, 32));
        mx = fmaxf(mx, __shfl_xor(mx, 8, 32));
        float mnew = fmaxf(mrow[v], mx);
        float a = __expf(mrow[v] - mnew);
        float p = __expf(sv[v] - mnew);
        float l = p;
        l += __shfl_xor(l, 1, 32);
        l += __shfl_xor(l, 2, 32);
        l += __shfl_xor(l, 4, 32);
        l += __shfl_xor(l, 8, 32);
        lrow[v] = lrow[v] * a + l;
        mrow[v] = mnew;
        alpha[v] = a;
        pv[v] = p;
      }

      // rescale existing output accumulator rows
#pragma unroll
      for (int nt = 0; nt < 4; nt++)
#pragma unroll
        for (int v = 0; v < 8; v++) accO[nt][v] *= alpha[v];

      // ---- C/D layout -> A layout for P, via per-wave LDS scratch ----
#pragma unroll
      for (int v = 0; v < 8; v++) Ps[wave][v + 8 * hi][nIdx] = pv[v];
      asm volatile("s_wait_dscnt 0x0" ::: "memory");  // order LDS W->R in-wave

      // ---- accO += P(16x16) . V(16x64) ----
#pragma unroll
      for (int kk = 0; kk < 16; kk += 4) {
        v2f aP;
        aP.x = Ps[wave][nIdx][kk + kb];
        aP.y = Ps[wave][nIdx][kk + kb + 1];
#pragma unroll
        for (int nt = 0; nt < 4; nt++) {
          v2f bV;
          bV.x = Vs[buf][sub + kk + kb][nt * 16 + nIdx];
          bV.y = Vs[buf][sub + kk + kb + 1][nt * 16 + nIdx];
          accO[nt] = WMMA_F32(aP, bV, accO[nt]);
        }
      }
    }
    __syncthreads();   // all waves done with buf before it is re-staged
  }

  // ---- normalize and store attention output [B,H,S,Hd] ----
#pragma unroll
  for (int v = 0; v < 8; v++) lrow[v] = 1.0f / lrow[v];
#pragma unroll
  for (int nt = 0; nt < 4; nt++)
#pragma unroll
    for (int v = 0; v < 8; v++) {
      int row = q0 + v + 8 * hi;
      int d = nt * 16 + nIdx;
      O[((size_t)bh * S + row) * HD + d] = accO[nt][v] * lrow[v];
    }
}

// ---------------------------------------------------------------------------
extern "C" void kernel_launch(void* const* d_in, const int* in_sizes, int n_in,
                              void* d_out, int out_size, void* d_ws, size_t ws_size,
                              hipStream_t stream)
{
  (void)in_sizes; (void)n_in; (void)out_size; (void)ws_size;

  const float*         x    = (const float*)d_in[0];
  const unsigned char* mask = (const unsigned char*)d_in[1];  // jnp bool = 1 byte
  const float* Wq = (const float*)d_in[2];
  const float* bq = (const float*)d_in[3];
  const float* Wk = (const float*)d_in[4];
  const float* bk = (const float*)d_in[5];
  const float* Wv = (const float*)d_in[6];
  const float* bv = (const float*)d_in[7];
  const float* Wo = (const float*)d_in[8];
  const float* bo = (const float*)d_in[9];

  const int Bb = 2, S = 2048, D = 512, H = 8, Hd = 64;
  const int M = Bb * S;  // 4096

  float* Qb = (float*)d_ws;                  // [B,H,S,Hd] each = 8 MB
  float* Kb = Qb + (size_t)M * D;
  float* Vb = Kb + (size_t)M * D;
  float* Ab = Vb + (size_t)M * D;            // attention output, head-major

  dim3 gg(M / TS, D / TS);                   // (64, 8)
  gemm_wmma_f32<<<gg, 128, 0, stream>>>(x,  Wq, bq, Qb, M, D, D, 0, 1, S, Hd, H);
  gemm_wmma_f32<<<gg, 128, 0, stream>>>(x,  Wk, bk, Kb, M, D, D, 0, 1, S, Hd, H);
  gemm_wmma_f32<<<gg, 128, 0, stream>>>(x,  Wv, bv, Vb, M, D, D, 0, 1, S, Hd, H);

  attn_wmma_f32<<<dim3(S / 64, Bb * H), 128, 0, stream>>>(Qb, Kb, Vb, mask, Ab, S, H);

  gemm_wmma_f32<<<gg, 128, 0, stream>>>(Ab, Wo, bo, (float*)d_out, M, D, D, 1, 0, S, Hd, H);
}